// ReduceLayer_fc2_33887291965658
// MI455X (gfx1250) — compile-verified
//
#include <hip/hip_runtime.h>
#include <hip/hip_bf16.h>

// Problem constants (from the reference)
#define D_OUT    4096
#define D_IN     11008
#define REMAINED 5504

typedef __attribute__((ext_vector_type(2))) float v2f;
typedef __attribute__((ext_vector_type(8))) float v8f;

// ---------------------------------------------------------------------------
// Kernel 1: zero the accumulation vector v (11008 f32) in workspace.
// ---------------------------------------------------------------------------
__global__ void zero_v_kernel(float* __restrict__ v) {
    int i = blockIdx.x * blockDim.x + threadIdx.x;
    if (i < D_IN) v[i] = 0.0f;
}

// ---------------------------------------------------------------------------
// Kernel 2: scatter-add  v[cols[k]] += x[k]  for the selected cluster row.
// Converts the gathered GEMV into a dense GEMV (duplicate indices handled
// exactly by the atomic sum). act_idx is int64 per the reference.
// ---------------------------------------------------------------------------
__global__ void scatter_v_kernel(const float* __restrict__ x,
                                 const long long* __restrict__ act_idx,
                                 const int* __restrict__ cluster,
                                 float* __restrict__ v) {
    int k = blockIdx.x * blockDim.x + threadIdx.x;
    if (k < REMAINED) {
        long long c = act_idx[(long long)(*cluster) * (long long)REMAINED + k];
        atomicAdd(v + c, x[k]);
    }
}

// ---------------------------------------------------------------------------
// Kernel 3: dense GEMV  out = W(4096x11008) * v + bias  using
// V_WMMA_F32_16X16X4_F32 as the reduction engine.
//
// Per wave: 16 output rows (M across lanes 0-15 / 16-31), K swept 4 at a time.
//   A tile (16x4 f32, 2 VGPRs): lane L holds W[mrow, kb+2*(L>>4) .. +1]  -> one float2 load
//   B tile (4x16  f32, 2 VGPRs): lane L needs v[kb+2*(L>>4) .. +1]      -> same float2 pattern
// All 16 columns of D are identical partial sums; lane 0 carries M=0..7,
// lane 16 carries M=8..15 in the 8 accumulator VGPRs (documented C/D layout).
// 8 waves per block split K (1376 each); LDS reduction + bias + store.
// ---------------------------------------------------------------------------
#define KWAVES 8

__global__ __launch_bounds__(256) void gemv_wmma_kernel(const float* __restrict__ W,
                                                        const float* __restrict__ bias,
                                                        const float* __restrict__ v,
                                                        float* __restrict__ out) {
    __shared__ float red[KWAVES][16];

    const int tile  = blockIdx.x;          // 0..255  (M tile of 16 rows)
    const int lane  = threadIdx.x & 31;
    const int wave  = threadIdx.x >> 5;    // 0..7    (K split)
    const int mrow  = tile * 16 + (lane & 15);
    const int khalf = lane >> 4;           // 0: K={0,1}, 1: K={2,3} within each 4-chunk

    const int kChunk = D_IN / KWAVES;      // 1376, multiple of 4
    const int k0 = wave * kChunk;
    const int k1 = k0 + kChunk;

    const float* wrow = W + (size_t)mrow * (size_t)D_IN;

    v8f acc = {};
    for (int kb = k0; kb < k1; kb += 4) {
        v2f a = *(const v2f*)(wrow + kb + 2 * khalf);   // A: 16x4 tile of W
        v2f b = *(const v2f*)(v    + kb + 2 * khalf);   // B: v chunk, all N columns equal
        // 8 args: (neg_a, A, neg_b, B, c_mod, C, reuse_a, reuse_b)
        acc = __builtin_amdgcn_wmma_f32_16x16x4_f32(
                  false, a, false, b, (short)0, acc, false, false);
    }

    // Column N=0 lives in lanes 0 (M=0..7) and 16 (M=8..15).
    if ((lane & 15) == 0) {
        const int mbase = khalf * 8;
        #pragma unroll
        for (int r = 0; r < 8; ++r) red[wave][mbase + r] = acc[r];
    }
    __syncthreads();

    if (threadIdx.x < 16) {
        float s = bias[tile * 16 + threadIdx.x];
        #pragma unroll
        for (int w = 0; w < KWAVES; ++w) s += red[w][threadIdx.x];
        out[tile * 16 + threadIdx.x] = s;
    }
}

// ---------------------------------------------------------------------------
// Launch: v lives in workspace (44 KB). Stream order guarantees
// zero -> scatter -> gemv dependencies.
// ---------------------------------------------------------------------------
extern "C" void kernel_launch(void* const* d_in, const int* in_sizes, int n_in,
                              void* d_out, int out_size, void* d_ws, size_t ws_size,
                              hipStream_t stream) {
    const float*     x       = (const float*)d_in[0];
    const float*     weight  = (const float*)d_in[1];
    const float*     bias    = (const float*)d_in[2];
    const long long* act_idx = (const long long*)d_in[3];
    const int*       cluster = (const int*)d_in[4];   // little-endian low word works for i32/i64
    float*           out     = (float*)d_out;

    float* v = (float*)d_ws;                          // 11008 f32 accumulation vector

    zero_v_kernel<<<(D_IN + 255) / 256, 256, 0, stream>>>(v);
    scatter_v_kernel<<<(REMAINED + 255) / 256, 256, 0, stream>>>(x, act_idx, cluster, v);
    gemv_wmma_kernel<<<D_OUT / 16, 256, 0, stream>>>(weight, bias, v, out);
}